// GNN_13709535609075
// MI455X (gfx1250) — compile-verified
//
#include <hip/hip_runtime.h>
#include <hip/hip_bf16.h>

typedef __attribute__((ext_vector_type(2))) float v2f;
typedef __attribute__((ext_vector_type(8))) float v8f;

#define C_IN 128

// ---------------------------------------------------------------------------
// Zero scratch region (agg + cnt) — grid-stride, deterministic each call.
// ---------------------------------------------------------------------------
__global__ void sage_zero(float* __restrict__ p, long long n) {
  long long i = (long long)blockIdx.x * blockDim.x + threadIdx.x;
  long long stride = (long long)gridDim.x * blockDim.x;
  for (; i < n; i += stride) p[i] = 0.0f;
}

// ---------------------------------------------------------------------------
// Edge scatter: one wave32 per edge. Lanes 0..31 cover the 128-float feature
// row as 32 float4 chunks: coalesced 512B read of x[src], coalesced burst of
// global_atomic_add_f32 into agg[dst] (L2-resident, resolved in L2 atomic
// units). Lane 0 also bumps the degree counter.
// ---------------------------------------------------------------------------
__global__ void sage_scatter(const float* __restrict__ x,
                             const int* __restrict__ src,
                             const int* __restrict__ dst,
                             float* __restrict__ agg,
                             float* __restrict__ cnt,
                             int nEdges) {
  long long t = (long long)blockIdx.x * blockDim.x + threadIdx.x;
  int e  = (int)(t >> 5);
  int cq = (int)(t & 31);
  if (e >= nEdges) return;
  int s = src[e];
  int d = dst[e];
  const float4* xrow = (const float4*)(x + (long long)s * C_IN);
  float4 v = xrow[cq];
  float* arow = agg + (long long)d * C_IN + cq * 4;
  atomicAdd(arow + 0, v.x);
  atomicAdd(arow + 1, v.y);
  atomicAdd(arow + 2, v.z);
  atomicAdd(arow + 3, v.w);
  if (cq == 0) atomicAdd(cnt + d, 1.0f);
}

// ---------------------------------------------------------------------------
// Fused normalize + dual-GEMM + bias (+ReLU):
//   out = (agg / max(cnt,1)) @ Wl + bl + x @ Wr
// One wave per 16x16 output tile, V_WMMA_F32_16X16X4_F32 over K=128 twice,
// accumulating in a single v8f. fp32-exact path (matches reference dtype).
//
// Fragment layouts (CDNA5 ISA 7.12.2, wave32):
//  A 16x4 f32: lane L holds row (L&15); vgpr0 = K = 2*(L>>4), vgpr1 = K+1.
//  B 4x16 f32: lane L holds col (L&15); vgpr0 = K = 2*(L>>4) row, vgpr1 = K+1.
//  C/D 16x16 f32: vgpr r -> row r + 8*(L>>4), col (L&15).
// ---------------------------------------------------------------------------
template <int COUT, bool RELU>
__global__ void sage_gemm(const float* __restrict__ xin,
                          const float* __restrict__ agg,
                          const float* __restrict__ cnt,
                          const float* __restrict__ Wl,
                          const float* __restrict__ bl,
                          const float* __restrict__ Wr,
                          float* __restrict__ out,
                          int totalTiles) {
  const int tilesN = COUT / 16;
  int wave = (int)(((long long)blockIdx.x * blockDim.x + threadIdx.x) >> 5);
  if (wave >= totalTiles) return;          // wave-uniform guard (EXEC stays all-1s)
  int lane = threadIdx.x & 31;
  int m0 = (wave / tilesN) * 16;
  int n0 = (wave % tilesN) * 16;
  int r  = lane & 15;                       // A-row / B-col within tile
  int kh = lane >> 4;                       // K half-select
  int col = n0 + r;

  const float* arow = agg + (long long)(m0 + r) * C_IN;
  const float* xrow = xin + (long long)(m0 + r) * C_IN;
  float inv = 1.0f / fmaxf(cnt[m0 + r], 1.0f);   // mean normalization, fused

  v8f acc = {};

  // agg_norm @ Wl
#pragma unroll
  for (int k = 0; k < C_IN; k += 4) {
    v2f a, b;
    int kk = k + 2 * kh;
    a[0] = arow[kk] * inv;
    a[1] = arow[kk + 1] * inv;
    b[0] = Wl[(long long)kk * COUT + col];
    b[1] = Wl[(long long)(kk + 1) * COUT + col];
    acc = __builtin_amdgcn_wmma_f32_16x16x4_f32(false, a, false, b,
                                                (short)0, acc, false, false);
  }
  // x @ Wr
#pragma unroll
  for (int k = 0; k < C_IN; k += 4) {
    v2f a, b;
    int kk = k + 2 * kh;
    a[0] = xrow[kk];
    a[1] = xrow[kk + 1];
    b[0] = Wr[(long long)kk * COUT + col];
    b[1] = Wr[(long long)(kk + 1) * COUT + col];
    acc = __builtin_amdgcn_wmma_f32_16x16x4_f32(false, a, false, b,
                                                (short)0, acc, false, false);
  }

  float bias = bl[col];
#pragma unroll
  for (int rr = 0; rr < 8; ++rr) {
    float v = acc[rr] + bias;
    if (RELU) v = fmaxf(v, 0.0f);
    out[(long long)(m0 + rr + 8 * kh) * COUT + col] = v;
  }
}

// ---------------------------------------------------------------------------
// Host-side orchestration (graph-capture safe: only kernel launches on stream)
// ---------------------------------------------------------------------------
extern "C" void kernel_launch(void* const* d_in, const int* in_sizes, int n_in,
                              void* d_out, int out_size, void* d_ws, size_t ws_size,
                              hipStream_t stream) {
  const float* x0  = (const float*)d_in[0];
  const int*   ei  = (const int*)d_in[1];     // edge_index [2, E] row-major
  const float* Wl0 = (const float*)d_in[2];
  const float* bl0 = (const float*)d_in[3];
  const float* Wr0 = (const float*)d_in[4];
  const float* Wl1 = (const float*)d_in[5];
  const float* bl1 = (const float*)d_in[6];
  const float* Wr1 = (const float*)d_in[7];
  const float* Wl2 = (const float*)d_in[8];
  const float* bl2 = (const float*)d_in[9];
  const float* Wr2 = (const float*)d_in[10];

  const int N = in_sizes[0] / C_IN;           // 100000 (exactly 6250 * 16)
  const int E = in_sizes[1] / 2;              // 1600000
  const int* src = ei;
  const int* dst = ei + E;

  // Workspace carve-up: h0, h1 (N x 128), agg (N x 128), cnt (N)
  float* h0  = (float*)d_ws;
  float* h1  = h0 + (size_t)N * C_IN;
  float* agg = h1 + (size_t)N * C_IN;
  float* cnt = agg + (size_t)N * C_IN;
  const long long aggCntElems = (long long)N * C_IN + N;

  const int ZB = 1024;                 // blocks for zeroing (grid-stride)
  const long long scatterThreads = (long long)E * 32;
  const int scatterBlocks = (int)((scatterThreads + 255) / 256);

  const int tilesM = (N + 15) / 16;

  // ---- Layer 0: 128 -> 128, ReLU ----
  sage_zero<<<ZB, 256, 0, stream>>>(agg, aggCntElems);
  sage_scatter<<<scatterBlocks, 256, 0, stream>>>(x0, src, dst, agg, cnt, E);
  {
    int tiles = tilesM * (128 / 16);
    int blocks = (tiles + 7) / 8;      // 8 waves per 256-thread block
    sage_gemm<128, true><<<blocks, 256, 0, stream>>>(x0, agg, cnt, Wl0, bl0, Wr0, h0, tiles);
  }

  // ---- Layer 1: 128 -> 128, ReLU ----
  sage_zero<<<ZB, 256, 0, stream>>>(agg, aggCntElems);
  sage_scatter<<<scatterBlocks, 256, 0, stream>>>(h0, src, dst, agg, cnt, E);
  {
    int tiles = tilesM * (128 / 16);
    int blocks = (tiles + 7) / 8;
    sage_gemm<128, true><<<blocks, 256, 0, stream>>>(h0, agg, cnt, Wl1, bl1, Wr1, h1, tiles);
  }

  // ---- Layer 2: 128 -> 64, no ReLU, write d_out ----
  sage_zero<<<ZB, 256, 0, stream>>>(agg, aggCntElems);
  sage_scatter<<<scatterBlocks, 256, 0, stream>>>(h1, src, dst, agg, cnt, E);
  {
    int tiles = tilesM * (64 / 16);
    int blocks = (tiles + 7) / 8;
    sage_gemm<64, false><<<blocks, 256, 0, stream>>>(h1, agg, cnt, Wl2, bl2, Wr2,
                                                     (float*)d_out, tiles);
  }
}